// DGCNNFilter_19387482374727
// MI455X (gfx1250) — compile-verified
//
#include <hip/hip_runtime.h>
#include <hip/hip_bf16.h>
#include <float.h>

typedef __attribute__((ext_vector_type(2))) float v2f;
typedef __attribute__((ext_vector_type(8))) float v8f;

// D = A(16x4 f32) * B(4x16 f32) + C(16x16 f32), wave32 WMMA.
#define WMMA_F32X4(a, b, c) \
  __builtin_amdgcn_wmma_f32_16x16x4_f32(false, (a), false, (b), (short)0, (c), false, false)

// ---------------------------------------------------------------------------
// Small prep kernels
// ---------------------------------------------------------------------------

// Pad [BN,3] -> [BN,4] (zero 4th feature) so layer-1 can use the same WMMA path.
__global__ void k_pad(const float* __restrict__ x, float* __restrict__ xp, int total) {
  int t = blockIdx.x * blockDim.x + threadIdx.x;
  if (t >= total) return;
  xp[t * 4 + 0] = x[t * 3 + 0];
  xp[t * 4 + 1] = x[t * 3 + 1];
  xp[t * 4 + 2] = x[t * 3 + 2];
  xp[t * 4 + 3] = 0.0f;
}

// Build combined weight Wc[Fp, C]: cols [0,o) = Wt - Wb, cols [o,2o) = Wb, rest 0.
// W is the reference [2*Fin, o] matrix.
__global__ void k_prepw(const float* __restrict__ W, float* __restrict__ Wc,
                        int Fin, int Fp, int o, int C) {
  int t = blockIdx.x * blockDim.x + threadIdx.x;
  if (t >= Fp * C) return;
  int kp = t / C, c = t % C;
  float v = 0.0f;
  if (kp < Fin) {
    if (c < o)            v = W[kp * o + c] - W[(kp + Fin) * o + c];
    else if (c < 2 * o)   v = W[(kp + Fin) * o + (c - o)];
  }
  Wc[t] = v;
}

// Per-point squared norms.
__global__ void k_sqnorm(const float* __restrict__ X, float* __restrict__ sq,
                         int total, int F) {
  int t = blockIdx.x * blockDim.x + threadIdx.x;
  if (t >= total) return;
  const float* p = X + (size_t)t * F;
  float s = 0.0f;
  for (int i = 0; i < F; ++i) { float v = p[i]; s += v * v; }
  sq[t] = s;
}

// ---------------------------------------------------------------------------
// Fused Gram-matrix tile (WMMA f32 16x16x4) + running top-32 selection.
// One wave per 16-row tile; sweeps all N columns in 16-wide tiles; never
// materializes the NxN distance matrix. Common case (no candidate beats the
// current 32nd-best of any row) is a barrier-free full-wave register test
// against per-row thresholds, resolved with a single wave ballot.
// ---------------------------------------------------------------------------
template <int F>
__global__ __launch_bounds__(32) void k_topk(const float* __restrict__ X,
                                             const float* __restrict__ sq,
                                             int* __restrict__ idx, int N) {
  constexpr int NS = F / 4;           // K-steps per WMMA accumulation chain
  const int b  = blockIdx.y;
  const int r0 = blockIdx.x * 16;
  const float* Xb  = X  + (size_t)b * N * F;
  const float* sqb = sq + (size_t)b * N;
  const int l  = threadIdx.x;
  const int lr = l & 15;
  const int hi = l >> 4;

  __shared__ float tile[16][16];
  __shared__ float bestD[16][32];
  __shared__ int   bestI[16][32];
  __shared__ float rowWorst[16];

  for (int i = l; i < 16 * 32; i += 32) {
    (&bestD[0][0])[i] = FLT_MAX;
    (&bestI[0][0])[i] = 0;
  }
  if (l < 16) rowWorst[l] = FLT_MAX;
  __syncthreads();

  // A fragments for rows r0..r0+15, held in registers for the entire sweep.
  v2f a[NS];
  {
    const float* pr = Xb + (size_t)(r0 + lr) * F + 2 * hi;
#pragma unroll
    for (int s = 0; s < NS; ++s) { a[s].x = pr[4 * s + 0]; a[s].y = pr[4 * s + 1]; }
  }
  float sqrow[8];
#pragma unroll
  for (int v = 0; v < 8; ++v) sqrow[v] = sqb[r0 + v + 8 * hi];

  // Per-row selection state (lanes 0..15 own one row each).
  int   cnt = 0, wsl = 0;
  float worst = FLT_MAX;

  const int NT = N / 16;
  for (int ct = 0; ct < NT; ++ct) {
    const int c0 = ct * 16;
    if (ct + 1 < NT)  // overlap next B-tile's cache fill with this tile's work
      __builtin_prefetch(Xb + (size_t)(c0 + 16 + lr) * F, 0, 1);

    v8f acc = {0.f, 0.f, 0.f, 0.f, 0.f, 0.f, 0.f, 0.f};
    const float* pc = Xb + (size_t)(c0 + lr) * F + 2 * hi;
#pragma unroll
    for (int s = 0; s < NS; ++s) {
      v2f bb; bb.x = pc[4 * s + 0]; bb.y = pc[4 * s + 1];
      acc = WMMA_F32X4(a[s], bb, acc);
    }
    const float sqc = sqb[c0 + lr];
    float dv[8];
#pragma unroll
    for (int v = 0; v < 8; ++v) dv[v] = sqrow[v] + sqc - 2.0f * acc[v];

    // Fast reject: each lane tests its 8 register distances against the
    // thresholds of its 8 rows (broadcast 32B LDS read); one ballot decides.
    bool need = true;
    if (ct >= 2) {                    // after warm-up all rows have 32 entries
      bool pass = false;
#pragma unroll
      for (int v = 0; v < 8; ++v) pass |= (dv[v] < rowWorst[8 * hi + v]);
      need = (__ballot(pass) != 0ull);
    }

    if (need) {
      __syncthreads();
#pragma unroll
      for (int v = 0; v < 8; ++v) tile[v + 8 * hi][lr] = dv[v];
      __syncthreads();
      if (l < 16) {
        const int m = l;
        for (int n = 0; n < 16; ++n) {
          const float d = tile[m][n];
          if (cnt < 32) {
            bestD[m][cnt] = d; bestI[m][cnt] = c0 + n;
            ++cnt;
            if (cnt == 32) {
              worst = -FLT_MAX;
              for (int s2 = 0; s2 < 32; ++s2) {
                float v = bestD[m][s2];
                if (v > worst) { worst = v; wsl = s2; }
              }
              rowWorst[m] = worst;
            }
          } else if (d < worst) {
            bestD[m][wsl] = d; bestI[m][wsl] = c0 + n;
            worst = -FLT_MAX;
            for (int s2 = 0; s2 < 32; ++s2) {
              float v = bestD[m][s2];
              if (v > worst) { worst = v; wsl = s2; }
            }
            rowWorst[m] = worst;
          }
        }
      }
      __syncthreads();
    }
  }
  __syncthreads();
  for (int i = l; i < 16 * 32; i += 32) {
    int m = i >> 5, s = i & 31;
    idx[((size_t)b * N + r0 + m) * 32 + s] = bestI[m][s];
  }
}

// ---------------------------------------------------------------------------
// Feature transform: Y[b] = X[b] @ Wc, [N,F] x [F,C], WMMA f32 16x16x4.
// ---------------------------------------------------------------------------
template <int F>
__global__ __launch_bounds__(32) void k_gemm(const float* __restrict__ X,
                                             const float* __restrict__ Wc,
                                             float* __restrict__ Y, int N, int C) {
  constexpr int NS = F / 4;
  const int b  = blockIdx.z;
  const int r0 = blockIdx.x * 16, c0 = blockIdx.y * 16;
  const int l  = threadIdx.x;
  const int lr = l & 15, hi = l >> 4;
  const float* Xb = X + (size_t)b * N * F;

  v8f acc = {0.f, 0.f, 0.f, 0.f, 0.f, 0.f, 0.f, 0.f};
  const float* pr = Xb + (size_t)(r0 + lr) * F + 2 * hi;
#pragma unroll
  for (int s = 0; s < NS; ++s) {
    v2f a; a.x = pr[4 * s + 0]; a.y = pr[4 * s + 1];
    const int k = 4 * s + 2 * hi;
    v2f bb;
    bb.x = Wc[(size_t)k * C + c0 + lr];
    bb.y = Wc[(size_t)(k + 1) * C + c0 + lr];
    acc = WMMA_F32X4(a, bb, acc);
  }
  float* Yb = Y + (size_t)b * N * C;
#pragma unroll
  for (int v = 0; v < 8; ++v) {
    Yb[(size_t)(r0 + v + 8 * hi) * C + c0 + lr] = acc[v];
  }
}

// ---------------------------------------------------------------------------
// h[n,c] = AY[n,c] + bias[c] + max_k AY[idx[n,k], o + c]
// ---------------------------------------------------------------------------
__global__ void k_gathermax(const float* __restrict__ AY, const int* __restrict__ idx,
                            const float* __restrict__ bias, float* __restrict__ out,
                            int N, int C, int o, int outStride, int totalBN) {
  int t = blockIdx.x * blockDim.x + threadIdx.x;
  if (t >= totalBN * o) return;
  const int c  = t % o;
  const int bn = t / o;
  const int b  = bn / N;
  const int n  = bn % N;
  const float* AYb = AY + (size_t)b * N * C;
  const int* id = idx + (size_t)bn * 32;
  float m = -FLT_MAX;
  for (int k = 0; k < 32; ++k) {
    float v = AYb[(size_t)id[k] * C + o + c];
    m = fmaxf(m, v);
  }
  out[(size_t)bn * outStride + c] = AYb[(size_t)n * C + c] + bias[c] + m;
}

// ---------------------------------------------------------------------------
// Training-mode BatchNorm over ReLU(h), deterministic two-stage reduction.
// ---------------------------------------------------------------------------
__global__ __launch_bounds__(256) void k_bnstats(const float* __restrict__ H,
                                                 float* __restrict__ part, int rows) {
  __shared__ float s1[4][64];
  __shared__ float s2s[4][64];
  const int c = threadIdx.x & 63;
  const int g = threadIdx.x >> 6;
  const int perBlk = rows / gridDim.x;
  const int start  = blockIdx.x * perBlk;
  float a = 0.0f, aa = 0.0f;
  for (int r = g; r < perBlk; r += 4) {
    float v = fmaxf(H[(size_t)(start + r) * 64 + c], 0.0f);
    a += v; aa += v * v;
  }
  s1[g][c] = a; s2s[g][c] = aa;
  __syncthreads();
  if (g == 0) {
    part[(size_t)blockIdx.x * 128 + c]      = s1[0][c] + s1[1][c] + s1[2][c] + s1[3][c];
    part[(size_t)blockIdx.x * 128 + 64 + c] = s2s[0][c] + s2s[1][c] + s2s[2][c] + s2s[3][c];
  }
}

__global__ void k_bnfinal(const float* __restrict__ part, const float* __restrict__ gamma,
                          const float* __restrict__ beta, float* __restrict__ ss,
                          int nblk, int rows) {
  int c = threadIdx.x;
  if (c >= 64) return;
  float s = 0.0f, s2 = 0.0f;
  for (int i = 0; i < nblk; ++i) { s += part[i * 128 + c]; s2 += part[i * 128 + 64 + c]; }
  const float inv  = 1.0f / (float)rows;
  const float mean = s * inv;
  const float var  = s2 * inv - mean * mean;
  const float scale = gamma[c] * rsqrtf(var + 1e-5f);
  ss[c]      = scale;
  ss[64 + c] = beta[c] - mean * scale;
}

__global__ void k_bnapply(const float* __restrict__ H, const float* __restrict__ ss,
                          float* __restrict__ Xn, int total) {
  int t = blockIdx.x * blockDim.x + threadIdx.x;
  if (t >= total) return;
  const int c = t & 63;
  const float v = fmaxf(H[t], 0.0f);
  Xn[t] = v * ss[c] + ss[64 + c];
}

// ---------------------------------------------------------------------------
// MSE loss (deterministic fixed-order reduction).
// ---------------------------------------------------------------------------
__global__ __launch_bounds__(256) void k_losspart(const float* __restrict__ h,
                                                  const float* __restrict__ tgt,
                                                  float* __restrict__ part, int total) {
  __shared__ float sh[256];
  const int tid = threadIdx.x;
  float a = 0.0f;
  for (int i = blockIdx.x * 256 + tid; i < total; i += gridDim.x * 256) {
    float d = h[i] - tgt[i];
    a += d * d;
  }
  sh[tid] = a;
  __syncthreads();
  for (int s = 128; s > 0; s >>= 1) {
    if (tid < s) sh[tid] += sh[tid + s];
    __syncthreads();
  }
  if (tid == 0) part[blockIdx.x] = sh[0];
}

__global__ void k_lossfinal(const float* __restrict__ part, float* __restrict__ out,
                            int nblk, float invTotal) {
  if (threadIdx.x == 0) {
    float s = 0.0f;
    for (int i = 0; i < nblk; ++i) s += part[i];
    *out = s * invTotal;
  }
}

// ---------------------------------------------------------------------------
extern "C" void kernel_launch(void* const* d_in, const int* in_sizes, int n_in,
                              void* d_out, int out_size, void* d_ws, size_t ws_size,
                              hipStream_t stream) {
  (void)in_sizes; (void)n_in; (void)out_size; (void)ws_size;
  const float* x      = (const float*)d_in[0];
  const float* target = (const float*)d_in[1];
  const float* W1 = (const float*)d_in[2];  const float* b1  = (const float*)d_in[3];
  const float* g1 = (const float*)d_in[4];  const float* be1 = (const float*)d_in[5];
  const float* W2 = (const float*)d_in[6];  const float* b2  = (const float*)d_in[7];
  const float* g2 = (const float*)d_in[8];  const float* be2 = (const float*)d_in[9];
  const float* W3 = (const float*)d_in[10]; const float* b3  = (const float*)d_in[11];
  float* out = (float*)d_out;

  const int B = 8, N = 4096, BN = B * N;

  // Workspace layout (floats / ints), ~47 MB total.
  float* ws  = (float*)d_ws;
  float* wc1 = ws;                               // 4*128
  float* wc2 = wc1 + 4 * 128;                    // 64*128
  float* wc3 = wc2 + 64 * 128;                   // 64*16
  float* sq  = wc3 + 64 * 16;                    // BN
  int*   idx = (int*)(sq + BN);                  // BN*32
  float* XA  = (float*)(idx + (size_t)BN * 32);  // BN*64 (layer1 input stride 4, layer3 input stride 64)
  float* XB2 = XA + (size_t)BN * 64;             // BN*64 (layer2 input)
  float* AY  = XB2 + (size_t)BN * 64;            // BN*128
  float* H   = AY + (size_t)BN * 128;            // BN*64 pre-activation
  float* bnp = H + (size_t)BN * 64;              // 128*128 BN partials
  float* ss  = bnp + 128 * 128;                  // 128 scale/shift
  float* lp  = ss + 128;                         // 128 loss partials

  // Weight prep + input padding.
  k_prepw<<<(4 * 128 + 255) / 256, 256, 0, stream>>>(W1, wc1, 3, 4, 64, 128);
  k_prepw<<<(64 * 128 + 255) / 256, 256, 0, stream>>>(W2, wc2, 64, 64, 64, 128);
  k_prepw<<<(64 * 16 + 255) / 256, 256, 0, stream>>>(W3, wc3, 64, 64, 3, 16);
  k_pad<<<(BN + 255) / 256, 256, 0, stream>>>(x, XA, BN);

  const dim3 gt(N / 16, B);
  const dim3 gg(N / 16, 128 / 16, B);
  const dim3 gg3(N / 16, 1, B);

  // ---- Layer 1 (F=4 padded, o=64) ----
  k_sqnorm<<<(BN + 255) / 256, 256, 0, stream>>>(XA, sq, BN, 4);
  k_topk<4><<<gt, 32, 0, stream>>>(XA, sq, idx, N);
  k_gemm<4><<<gg, 32, 0, stream>>>(XA, wc1, AY, N, 128);
  k_gathermax<<<(BN * 64 + 255) / 256, 256, 0, stream>>>(AY, idx, b1, H, N, 128, 64, 64, BN);
  k_bnstats<<<128, 256, 0, stream>>>(H, bnp, BN);
  k_bnfinal<<<1, 64, 0, stream>>>(bnp, g1, be1, ss, 128, BN);
  k_bnapply<<<(BN * 64 + 255) / 256, 256, 0, stream>>>(H, ss, XB2, BN * 64);

  // ---- Layer 2 (F=64, o=64) ----
  k_sqnorm<<<(BN + 255) / 256, 256, 0, stream>>>(XB2, sq, BN, 64);
  k_topk<64><<<gt, 32, 0, stream>>>(XB2, sq, idx, N);
  k_gemm<64><<<gg, 32, 0, stream>>>(XB2, wc2, AY, N, 128);
  k_gathermax<<<(BN * 64 + 255) / 256, 256, 0, stream>>>(AY, idx, b2, H, N, 128, 64, 64, BN);
  k_bnstats<<<128, 256, 0, stream>>>(H, bnp, BN);
  k_bnfinal<<<1, 64, 0, stream>>>(bnp, g2, be2, ss, 128, BN);
  k_bnapply<<<(BN * 64 + 255) / 256, 256, 0, stream>>>(H, ss, XA, BN * 64);

  // ---- Layer 3 (F=64, o=3, C padded to 16), h -> d_out ----
  k_sqnorm<<<(BN + 255) / 256, 256, 0, stream>>>(XA, sq, BN, 64);
  k_topk<64><<<gt, 32, 0, stream>>>(XA, sq, idx, N);
  k_gemm<64><<<gg3, 32, 0, stream>>>(XA, wc3, AY, N, 16);
  k_gathermax<<<(BN * 3 + 255) / 256, 256, 0, stream>>>(AY, idx, b3, out, N, 16, 3, 3, BN);

  // ---- Loss -> d_out[BN*3] ----
  k_losspart<<<128, 256, 0, stream>>>(out, target, lp, BN * 3);
  k_lossfinal<<<1, 32, 0, stream>>>(lp, out + (size_t)BN * 3, 128, 1.0f / (float)(BN * 3));
}